// HashGridEncoder_31559419691689
// MI455X (gfx1250) — compile-verified
//
#include <hip/hip_runtime.h>
#include <cmath>

// ---------------------------------------------------------------------------
// Instant-NGP style multiresolution hash-grid encoder for gfx1250 (MI455X).
// One thread = one point. 16 levels x 8 corner gathers (float2, fits in L2),
// trilinear lerp tree in registers, non-temporal b128 output stores.
// xyz is staged into LDS with CDNA5 async global->LDS loads when available.
// ---------------------------------------------------------------------------

namespace {
constexpr int      kL     = 16;
constexpr int      kLog2T = 19;
constexpr int      kT     = 1 << kLog2T;
constexpr unsigned kMask  = (1u << 19) - 1u;
constexpr unsigned kP2    = 2654435761u;
constexpr unsigned kP3    = 805459861u;
constexpr int      kBlock = 256;

struct ResPack { int r[kL]; };
} // namespace

typedef __attribute__((address_space(1))) int GlobalInt;
typedef __attribute__((address_space(3))) int LdsInt;
typedef __attribute__((ext_vector_type(4))) float v4f;

#if defined(__HIP_DEVICE_COMPILE__) && defined(__gfx1250__) &&                 \
    __has_builtin(__builtin_amdgcn_global_load_async_to_lds_b32) &&            \
    __has_builtin(__builtin_amdgcn_s_wait_asynccnt)
#define USE_ASYNC_LDS 1
#else
#define USE_ASYNC_LDS 0
#endif

#if USE_ASYNC_LDS
__device__ __forceinline__ void async_g2lds_b32(const float* g, float* l) {
  // generic->AS casts via integer detour (generic LDS ptr low 32 bits = LDS offset)
  GlobalInt* gp = (GlobalInt*)(unsigned long long)g;
  LdsInt*    lp = (LdsInt*)(unsigned)(unsigned long long)l;
  __builtin_amdgcn_global_load_async_to_lds_b32(gp, lp, 0, 0);
}
#endif

__device__ __forceinline__ float2 lerp2(float2 a, float2 b, float t) {
  return make_float2(fmaf(t, b.x - a.x, a.x), fmaf(t, b.y - a.y, a.y));
}

__global__ __launch_bounds__(kBlock) void hashgrid_enc_kernel(
    const float* __restrict__ xyz, const float* __restrict__ tables,
    float* __restrict__ out, ResPack rp, int npts) {
  __shared__ float sxyz[kBlock * 3];

  const int       tid  = threadIdx.x;
  const long long base = (long long)blockIdx.x * kBlock;

  float x, y, z;
  if (base + kBlock <= npts) {
    // Full block: coalesced slab load (768 dwords) staged through LDS.
    const float* gsrc = xyz + base * 3;
#if USE_ASYNC_LDS
    async_g2lds_b32(gsrc + tid,              &sxyz[tid]);
    async_g2lds_b32(gsrc + kBlock + tid,     &sxyz[kBlock + tid]);
    async_g2lds_b32(gsrc + 2 * kBlock + tid, &sxyz[2 * kBlock + tid]);
    __builtin_amdgcn_s_wait_asynccnt(0);  // my wave's async writes landed in LDS
    __syncthreads();                      // all waves' writes visible
#else
    sxyz[tid]              = gsrc[tid];
    sxyz[kBlock + tid]     = gsrc[kBlock + tid];
    sxyz[2 * kBlock + tid] = gsrc[2 * kBlock + tid];
    __syncthreads();
#endif
    x = sxyz[tid * 3 + 0];
    y = sxyz[tid * 3 + 1];
    z = sxyz[tid * 3 + 2];
  } else {
    // Tail block (uniform branch per block): guarded direct loads.
    const long long pt = base + tid;
    if (pt < npts) {
      x = xyz[pt * 3 + 0];
      y = xyz[pt * 3 + 1];
      z = xyz[pt * 3 + 2];
    } else {
      x = y = z = 0.0f;
    }
  }

  const long long p = base + tid;
  if (p >= npts) return;

  const float2* __restrict__ tb2 = (const float2*)tables;
  float2 acc[kL];

#pragma unroll
  for (int l = 0; l < kL; ++l) {
    const float resf = (float)rp.r[l];
    // cell = 1/res; t = x/cell = x*res; d = frac(t)
    const float tx = x * resf, ty = y * resf, tz = z * resf;
    const float fx = floorf(tx), fy = floorf(ty), fz = floorf(tz);
    const float dx = tx - fx, dy = ty - fy, dz = tz - fz;

    const unsigned ix = (unsigned)(int)fx;
    const unsigned iy = (unsigned)(int)fy;
    const unsigned iz = (unsigned)(int)fz;

    // h = (ix*1 ^ iy*P2 ^ iz*P3) & mask; (i+1)*P = i*P + P (mod 2^32)
    const unsigned hx0 = ix,       hx1 = ix + 1u;
    const unsigned hy0 = iy * kP2, hy1 = hy0 + kP2;
    const unsigned hz0 = iz * kP3, hz1 = hz0 + kP3;

    const float2* __restrict__ lt = tb2 + (size_t)l * kT;
    // Vertex j -> (x = j>>2, y = (j>>1)&1, z = j&1)
    const float2 e0 = lt[(hx0 ^ hy0 ^ hz0) & kMask];  // 000
    const float2 e1 = lt[(hx0 ^ hy0 ^ hz1) & kMask];  // 001 (+z)
    const float2 e2 = lt[(hx0 ^ hy1 ^ hz0) & kMask];  // 010 (+y)
    const float2 e3 = lt[(hx0 ^ hy1 ^ hz1) & kMask];  // 011
    const float2 e4 = lt[(hx1 ^ hy0 ^ hz0) & kMask];  // 100 (+x)
    const float2 e5 = lt[(hx1 ^ hy0 ^ hz1) & kMask];  // 101
    const float2 e6 = lt[(hx1 ^ hy1 ^ hz0) & kMask];  // 110
    const float2 e7 = lt[(hx1 ^ hy1 ^ hz1) & kMask];  // 111

    const float2 c00 = lerp2(e0, e4, dx);
    const float2 c01 = lerp2(e1, e5, dx);
    const float2 c10 = lerp2(e2, e6, dx);
    const float2 c11 = lerp2(e3, e7, dx);
    const float2 c0  = lerp2(c00, c10, dy);
    const float2 c1  = lerp2(c01, c11, dy);
    acc[l]           = lerp2(c0, c1, dz);
  }

  // 128B contiguous row per point; non-temporal so the 128MB output stream
  // does not evict the 64MB table working set from the 192MB L2.
  v4f* o4 = (v4f*)(out + (size_t)p * (kL * 2));
#pragma unroll
  for (int k = 0; k < kL / 2; ++k) {
    v4f v;
    v.x = acc[2 * k].x;
    v.y = acc[2 * k].y;
    v.z = acc[2 * k + 1].x;
    v.w = acc[2 * k + 1].y;
    __builtin_nontemporal_store(v, o4 + k);
  }
}

extern "C" void kernel_launch(void* const* d_in, const int* in_sizes, int n_in,
                              void* d_out, int out_size, void* d_ws,
                              size_t ws_size, hipStream_t stream) {
  const float* xyz    = (const float*)d_in[0];
  const float* tables = (const float*)d_in[1];
  float*       out    = (float*)d_out;

  const int npts = in_sizes[0] / 3;

  // Replicate the reference's double-precision resolution schedule exactly:
  // B = exp((ln 512 - ln 16)/15); res_i = floor(16 * B^i)
  ResPack rp;
  const double B = std::exp((std::log(512.0) - std::log(16.0)) / 15.0);
  for (int i = 0; i < kL; ++i)
    rp.r[i] = (int)std::floor(16.0 * std::pow(B, (double)i));

  const int grid = (npts + kBlock - 1) / kBlock;
  hipLaunchKernelGGL(hashgrid_enc_kernel, dim3(grid), dim3(kBlock), 0, stream,
                     xyz, tables, out, rp, npts);
}